// SequentialPredictor_55035710931323
// MI455X (gfx1250) — compile-verified
//
#include <hip/hip_runtime.h>

typedef float v2f __attribute__((ext_vector_type(2)));
typedef float v8f __attribute__((ext_vector_type(8)));

#define HH 160
#define WW 160
#define HW (160 * 160)
#define CIN 256
#define OC_LOC 34          // NUM_CHARS + 2
#define VOC2 40            // VOC + 2
#define TSTEPS 25

static constexpr long RECS_ELEMS = 2L * 38 * HH * WW * TSTEPS;   // 48,640,000
static constexpr long MASKS_OFF  = RECS_ELEMS;                   // 48,640,000
static constexpr long LOCS_OFF   = MASKS_OFF + 2L * 32 * HH * WW;// 50,278,400

__device__ __forceinline__ int iclamp(int v, int lo, int hi) {
  return v < lo ? lo : (v > hi ? hi : v);
}

__device__ __forceinline__ long recs_idx(int n, int o, int h, int w, int t) {
  return (((long)(n * 38 + o) * HH + h) * WW + w) * TSTEPS + t;
}

// ---------------------------------------------------------------------------
// Kernel A: 3x3 SAME conv (256 -> 34). Channels 0,1 (+bias) -> locations ch 0,1
// (they equal the t=0 scan output since delta starts at zero). Channels 2..33
// (+bias) -> masks.
// ---------------------------------------------------------------------------
__global__ __launch_bounds__(256) void conv_init_kernel(
    const float* __restrict__ x, const float* __restrict__ lw,
    const float* __restrict__ lb, float* __restrict__ out) {
  __shared__ float xs[18][19];  // 16x16 tile + 1 halo, padded stride
  const int tid = threadIdx.x;
  const int b = blockIdx.x;           // 2 * 10 * 10 blocks
  const int n = b / 100;
  const int rem = b % 100;
  const int th0 = (rem / 10) * 16, tw0 = (rem % 10) * 16;
  const int ty = tid >> 4, tx = tid & 15;
  const int h = th0 + ty, w = tw0 + tx;

  float acc[OC_LOC];
#pragma unroll
  for (int o = 0; o < OC_LOC; ++o) acc[o] = 0.f;

  for (int c = 0; c < CIN; ++c) {
    // stage 18x18 x-patch (zero padded) for this channel
    for (int i = tid; i < 18 * 18; i += 256) {
      const int r = i / 18, cc = i % 18;
      const int hh = th0 + r - 1, ww2 = tw0 + cc - 1;
      float v = 0.f;
      if (hh >= 0 && hh < HH && ww2 >= 0 && ww2 < WW)
        v = x[((long)(n * CIN + c) * HH + hh) * WW + ww2];
      xs[r][cc] = v;
    }
    __syncthreads();

    float v[9];
#pragma unroll
    for (int ky = 0; ky < 3; ++ky)
#pragma unroll
      for (int kx = 0; kx < 3; ++kx) v[ky * 3 + kx] = xs[ty + ky][tx + kx];

    const float* wp = lw + (long)c * 9;  // lw layout [34][256][3][3]
#pragma unroll
    for (int o = 0; o < OC_LOC; ++o) {
      const float* wo = wp + (long)o * CIN * 9;  // wave-uniform -> scalar loads
#pragma unroll
      for (int k = 0; k < 9; ++k) acc[o] = fmaf(v[k], wo[k], acc[o]);
    }
    __syncthreads();
  }

#pragma unroll
  for (int o = 0; o < OC_LOC; ++o) {
    const float val = acc[o] + lb[o];
    long idx;
    if (o < 2)
      idx = LOCS_OFF + ((long)(n * 50 + o) * HH + h) * WW + w;
    else
      idx = MASKS_OFF + ((long)(n * 32 + (o - 2)) * HH + h) * WW + w;
    out[idx] = val;
  }
}

// ---------------------------------------------------------------------------
// Kernel B: 25-step per-pixel recurrence. One block = one image row (160 px),
// 10 waves x 16 pixels. Per step: bilinear-gather 256 channels of y, then
// [16px x 256] x [256 x 48] via V_WMMA_F32_16X16X4_F32 (3 N-tiles, 64 K-chunks).
// Outputs 0..37 (+bias) -> recs; 38,39 (+bias) -> delta via LDS.
// ---------------------------------------------------------------------------
__global__ __launch_bounds__(320) void seq_step_kernel(
    const float* __restrict__ y, const float* __restrict__ clf_w,
    const float* __restrict__ clf_b, float* __restrict__ out) {
  __shared__ v2f w1p[128][48];     // K-pair-major w1^T, padded to 48 outs
  __shared__ float dLds[WW][2];    // per-pixel delta exchange

  const int tid = threadIdx.x;

  // stage B operand: w1p[kp][o] = (w1[o][2kp], w1[o][2kp+1])
  for (int i = tid; i < 128 * 48; i += 320) {
    const int kp = i / 48, o = i % 48;
    v2f v;
    if (o < VOC2) {
      v.x = clf_w[(long)o * CIN + 2 * kp];
      v.y = clf_w[(long)o * CIN + 2 * kp + 1];
    } else {
      v.x = 0.f;
      v.y = 0.f;
    }
    w1p[kp][o] = v;
  }
  __syncthreads();

  const int bid = blockIdx.x;  // N * H = 320
  const int n = bid / HH, h = bid % HH;
  const int wave = tid >> 5, lane = tid & 31;
  const int m = lane & 15;      // pixel-in-wave (A rows) / out-col-in-tile
  const int khalf = lane >> 4;  // K half per 16x4 f32 A/B layout
  const int wp = wave * 16 + m; // pixel column in row

  const float* ybase = y + (long)n * CIN * HW;

  float bias[3];
#pragma unroll
  for (int t3 = 0; t3 < 3; ++t3) {
    const int o = t3 * 16 + m;
    bias[t3] = (o < VOC2) ? clf_b[o] : 0.f;
  }

  // initial location (written by conv kernel into locations ch 0,1)
  float ly = out[LOCS_OFF + ((long)(n * 50 + 0) * HH + h) * WW + wp];
  float lx = out[LOCS_OFF + ((long)(n * 50 + 1) * HH + h) * WW + wp];
  float dly = 0.f, dlx = 0.f;

  for (int t = 0; t < TSTEPS; ++t) {
    ly += dly;
    lx += dlx;

    // locations[n, 2t+j, h, wp]: lanes 0-15 write j=0 (y), 16-31 write j=1 (x)
    out[LOCS_OFF + ((long)(n * 50 + 2 * t + khalf) * HH + h) * WW + wp] =
        khalf ? lx : ly;

    // bilinear sampling parameters (uniform across channels)
    const float py = (float)h + ly;
    const float px = (float)wp + lx;
    const float fy = floorf(py), fx = floorf(px);
    const float tyf = py - fy, txf = px - fx;
    const int y0 = (int)fy, x0 = (int)fx;
    const int y1 = y0 + 1, x1 = x0 + 1;
    const float v00 = (y0 >= 0 && y0 < HH && x0 >= 0 && x0 < WW) ? 1.f : 0.f;
    const float v01 = (y0 >= 0 && y0 < HH && x1 >= 0 && x1 < WW) ? 1.f : 0.f;
    const float v10 = (y1 >= 0 && y1 < HH && x0 >= 0 && x0 < WW) ? 1.f : 0.f;
    const float v11 = (y1 >= 0 && y1 < HH && x1 >= 0 && x1 < WW) ? 1.f : 0.f;
    const int cy0 = iclamp(y0, 0, HH - 1), cy1 = iclamp(y1, 0, HH - 1);
    const int cx0 = iclamp(x0, 0, WW - 1), cx1 = iclamp(x1, 0, WW - 1);
    const int off00 = cy0 * WW + cx0, off01 = cy0 * WW + cx1;
    const int off10 = cy1 * WW + cx0, off11 = cy1 * WW + cx1;
    const float w00 = (1.f - tyf) * (1.f - txf) * v00;
    const float w01 = (1.f - tyf) * txf * v01;
    const float w10 = tyf * (1.f - txf) * v10;
    const float w11 = tyf * txf * v11;

    v8f acc0 = {}, acc1 = {}, acc2 = {};
    for (int k0 = 0; k0 < CIN; k0 += 4) {
      const int ca = k0 + 2 * khalf;  // A layout: lanes 0-15 -> K 0,1; 16-31 -> K 2,3
      const float* p0 = ybase + (long)ca * HW;
      const float* p1 = p0 + HW;
      v2f a;
      a.x = p0[off00] * w00 + p0[off01] * w01 + p0[off10] * w10 + p0[off11] * w11;
      a.y = p1[off00] * w00 + p1[off01] * w01 + p1[off10] * w10 + p1[off11] * w11;
      const int kp = (k0 >> 1) + khalf;
      const v2f b0 = w1p[kp][m];
      const v2f b1 = w1p[kp][16 + m];
      const v2f b2 = w1p[kp][32 + m];
      acc0 = __builtin_amdgcn_wmma_f32_16x16x4_f32(false, a, false, b0,
                                                   (short)0, acc0, false, false);
      acc1 = __builtin_amdgcn_wmma_f32_16x16x4_f32(false, a, false, b1,
                                                   (short)0, acc1, false, false);
      acc2 = __builtin_amdgcn_wmma_f32_16x16x4_f32(false, a, false, b2,
                                                   (short)0, acc2, false, false);
    }

    // delta (outputs 38,39) -> LDS. D layout: lane holds col o = 32 + m of
    // tile 2, rows (pixels) r + 8*khalf.
    if (m == 6 || m == 7) {
      const int j = m - 6;
      const float bb = clf_b[38 + j];
#pragma unroll
      for (int r = 0; r < 8; ++r) {
        const int p = wave * 16 + r + 8 * khalf;
        dLds[p][j] = acc2[r] + bb;
      }
    }
    __syncthreads();
    dly = dLds[wp][0];
    dlx = dLds[wp][1];

    // recs[n, o, h, p, t] for o < 38
#pragma unroll
    for (int t3 = 0; t3 < 3; ++t3) {
      const int o = t3 * 16 + m;
      if (o < 38) {
        const v8f av = (t3 == 0) ? acc0 : ((t3 == 1) ? acc1 : acc2);
        const float bo = bias[t3];
#pragma unroll
        for (int r = 0; r < 8; ++r) {
          const int p = wave * 16 + r + 8 * khalf;
          out[recs_idx(n, o, h, p, t)] = av[r] + bo;
        }
      }
    }
    __syncthreads();  // protect dLds before next step's writes
  }
}

extern "C" void kernel_launch(void* const* d_in, const int* in_sizes, int n_in,
                              void* d_out, int out_size, void* d_ws,
                              size_t ws_size, hipStream_t stream) {
  (void)in_sizes;
  (void)n_in;
  (void)out_size;
  (void)d_ws;
  (void)ws_size;
  const float* x = (const float*)d_in[0];
  const float* y = (const float*)d_in[1];
  const float* lw = (const float*)d_in[2];
  const float* lb = (const float*)d_in[3];
  const float* cw = (const float*)d_in[4];
  const float* cb = (const float*)d_in[5];
  float* out = (float*)d_out;

  conv_init_kernel<<<dim3(200), dim3(256), 0, stream>>>(x, lw, lb, out);
  seq_step_kernel<<<dim3(320), dim3(320), 0, stream>>>(y, cw, cb, out);
}